// SlotAttention_37245956390967
// MI455X (gfx1250) — compile-verified
//
#include <hip/hip_runtime.h>

typedef __attribute__((ext_vector_type(2))) float v2f;
typedef __attribute__((ext_vector_type(8))) float v8f;

#define N_IN      16384
#define D_IN      768
#define H_DIM     256
#define K_SLOTS   16
#define D_MLP     512
#define N_TILES   (N_IN / 16)
#define UP_CHUNKS 64
#define EPS_ATTN  1e-8f
#define LN_EPS    1e-5f

__device__ __forceinline__ v8f wmma4(v2f a, v2f b, v8f c) {
    // D = A(16x4 f32) * B(4x16 f32) + C(16x16 f32)
    return __builtin_amdgcn_wmma_f32_16x16x4_f32(false, a, false, b, (short)0, c, false, false);
}

__device__ __forceinline__ float sigmoidf_(float x) { return 1.0f / (1.0f + expf(-x)); }

// ---------------------------------------------------------------------------
// Kernel 1: x_ln = LN(inputs); k = x_ln @ Wk.T ; v = x_ln @ Wv.T
// grid = 1024 (16-row tiles of x), block = 256 (8 waves)
// ---------------------------------------------------------------------------
#define LDA_X 772  // 768 + 4 pad -> bank = (row*4 + d) % 64, conflict-free
__global__ void k_ln_kv(const float* __restrict__ x,
                        const float* __restrict__ lnw, const float* __restrict__ lnb,
                        const float* __restrict__ Wk, const float* __restrict__ Wv,
                        float* __restrict__ kout, float* __restrict__ vout) {
    __shared__ float xs[16 * LDA_X];
    const int tid = threadIdx.x;
    const int nb = blockIdx.x * 16;

    // ---- Phase 1: LayerNorm 16 rows; 16 threads per row (same half-wave) ----
    {
        const int row = tid >> 4;          // 0..15
        const int j   = tid & 15;
        const float* xr = x + (size_t)(nb + row) * D_IN;
        float s = 0.f, sq = 0.f;
        for (int d = j; d < D_IN; d += 16) { float t = xr[d]; s += t; sq += t * t; }
        for (int mk = 8; mk >= 1; mk >>= 1) { s += __shfl_xor(s, mk, 16); sq += __shfl_xor(sq, mk, 16); }
        const float mu = s * (1.0f / D_IN);
        const float rs = rsqrtf(sq * (1.0f / D_IN) - mu * mu + LN_EPS);
        for (int d = j; d < D_IN; d += 16)
            xs[row * LDA_X + d] = (xr[d] - mu) * rs * lnw[d] + lnb[d];
    }
    __syncthreads();

    // ---- Phase 2: WMMA; wave w handles 4 of 32 output tiles (16 k + 16 v) ----
    const int wave = tid >> 5;
    const int lane = tid & 31;
    const int half = lane >> 4;
    const int l16  = lane & 15;
    for (int t = 0; t < 4; ++t) {
        const int tile = wave * 4 + t;                 // 0..31
        const float* W   = (tile < 16) ? Wk : Wv;      // [256, 768] row-major
        float*       out = (tile < 16) ? kout : vout;  // [N, 256]
        const int h = (tile & 15) * 16 + l16;
        v8f c = {0.f, 0.f, 0.f, 0.f, 0.f, 0.f, 0.f, 0.f};
        const float* wr = W + (size_t)h * D_IN;
        for (int d = 0; d < D_IN; d += 4) {
            const int ko = d + 2 * half;
            v2f a; a.x = xs[l16 * LDA_X + ko]; a.y = xs[l16 * LDA_X + ko + 1];
            v2f b; b.x = wr[ko];               b.y = wr[ko + 1];
            c = wmma4(a, b, c);
        }
        #pragma unroll
        for (int j = 0; j < 8; ++j)
            out[(size_t)(nb + j + 8 * half) * H_DIM + h] = c[j];
    }
}

// ---------------------------------------------------------------------------
// Kernel: copy slots_init -> slots workspace
// ---------------------------------------------------------------------------
__global__ void k_init(const float* __restrict__ src, float* __restrict__ dst) {
    for (int i = threadIdx.x; i < K_SLOTS * H_DIM; i += blockDim.x) dst[i] = src[i];
}

// ---------------------------------------------------------------------------
// Kernel: q = LN(slots) @ Wq.T * scale    (1 block, 512 threads = 16 waves)
// ---------------------------------------------------------------------------
#define LDA_S 260  // 256 + 4 pad
__global__ void k_q(const float* __restrict__ slots,
                    const float* __restrict__ lnw, const float* __restrict__ lnb,
                    const float* __restrict__ Wq, float* __restrict__ qout) {
    __shared__ float ls[16 * LDA_S];
    const int tid = threadIdx.x;
    const int wave = tid >> 5;   // 0..15 (one wave per slot row / per h-tile)
    const int lane = tid & 31;

    // LN of slot row `wave`
    {
        const float* sr = slots + wave * H_DIM;
        float s = 0.f, sq = 0.f;
        for (int d = lane; d < H_DIM; d += 32) { float t = sr[d]; s += t; sq += t * t; }
        for (int mk = 16; mk >= 1; mk >>= 1) { s += __shfl_xor(s, mk, 32); sq += __shfl_xor(sq, mk, 32); }
        const float mu = s * (1.0f / H_DIM);
        const float rs = rsqrtf(sq * (1.0f / H_DIM) - mu * mu + LN_EPS);
        for (int d = lane; d < H_DIM; d += 32)
            ls[wave * LDA_S + d] = (sr[d] - mu) * rs * lnw[d] + lnb[d];
    }
    __syncthreads();

    // WMMA: wave w computes q[:, w*16 : w*16+16]
    const int half = lane >> 4, l16 = lane & 15;
    const int h = wave * 16 + l16;
    v8f c = {0.f, 0.f, 0.f, 0.f, 0.f, 0.f, 0.f, 0.f};
    const float* wr = Wq + (size_t)h * H_DIM;   // Wq: [256, 256] row-major
    for (int d = 0; d < H_DIM; d += 4) {
        const int ko = d + 2 * half;
        v2f a; a.x = ls[l16 * LDA_S + ko]; a.y = ls[l16 * LDA_S + ko + 1];
        v2f b; b.x = wr[ko];               b.y = wr[ko + 1];
        c = wmma4(a, b, c);
    }
    const float scale = 0.0625f;  // 256^-0.5
    #pragma unroll
    for (int j = 0; j < 8; ++j)
        qout[(j + 8 * half) * H_DIM + h] = c[j] * scale;
}

// ---------------------------------------------------------------------------
// Kernel: logits = k @ q.T (16x16 tile), softmax over slots, + eps,
// store attn_unnorm [N,16]; per-block partial column sums & sum-of-squares.
// grid = 1024, block = 32 (one wave)
// ---------------------------------------------------------------------------
__global__ void k_attn(const float* __restrict__ kmat, const float* __restrict__ q,
                       float* __restrict__ attn,
                       float* __restrict__ cs_part, float* __restrict__ sq_part) {
    const int lane = threadIdx.x;
    const int half = lane >> 4, l16 = lane & 15;
    const int nb = blockIdx.x * 16;

    v8f c = {0.f, 0.f, 0.f, 0.f, 0.f, 0.f, 0.f, 0.f};
    const float* kr = kmat + (size_t)(nb + l16) * H_DIM;  // A row (input n)
    const float* qr = q + (size_t)l16 * H_DIM;            // B col (slot)
    for (int d = 0; d < H_DIM; d += 4) {
        const int ko = d + 2 * half;
        v2f a; a.x = kr[ko]; a.y = kr[ko + 1];
        v2f b; b.x = qr[ko]; b.y = qr[ko + 1];
        c = wmma4(a, b, c);
    }

    // softmax across the 16 lanes holding one row's slot values
    float psum = 0.f, psq = 0.f;
    #pragma unroll
    for (int j = 0; j < 8; ++j) {
        float x = c[j];
        float m = x;
        for (int mk = 8; mk >= 1; mk >>= 1) m = fmaxf(m, __shfl_xor(m, mk, 16));
        float e = expf(x - m);
        float s = e;
        for (int mk = 8; mk >= 1; mk >>= 1) s += __shfl_xor(s, mk, 16);
        float p = e / s + EPS_ATTN;
        attn[(size_t)(nb + j + 8 * half) * K_SLOTS + l16] = p;
        psum += p; psq += p * p;
    }
    psum += __shfl_xor(psum, 16, 32);
    psq  += __shfl_xor(psq, 16, 32);
    if (lane < 16) {
        cs_part[blockIdx.x * K_SLOTS + lane] = psum;
        sq_part[blockIdx.x * K_SLOTS + lane] = psq;
    }
}

// Deterministic reduction of per-block partials (32 threads, 1 block)
__global__ void k_reduce_cs(const float* __restrict__ cs_part, const float* __restrict__ sq_part,
                            float* __restrict__ colsum, float* __restrict__ sqsum) {
    const int t = threadIdx.x;
    if (t < 16) {
        float s = 0.f;
        for (int b = 0; b < N_TILES; ++b) s += cs_part[b * K_SLOTS + t];
        colsum[t] = s;
    } else if (t < 32) {
        const int c0 = t - 16;
        float s = 0.f;
        for (int b = 0; b < N_TILES; ++b) s += sq_part[b * K_SLOTS + c0];
        sqsum[c0] = s;
    }
}

// ---------------------------------------------------------------------------
// Kernel: updates_raw = attn_unnorm.T @ v, K split into 64 chunks of 256 rows.
// grid = (64 chunks, 16 h-tiles), block = 32
// ---------------------------------------------------------------------------
__global__ void k_updates(const float* __restrict__ attn, const float* __restrict__ vmat,
                          float* __restrict__ up_part) {
    const int lane = threadIdx.x;
    const int half = lane >> 4, l16 = lane & 15;
    const int chunk = blockIdx.x;        // 0..63
    const int h = blockIdx.y * 16 + l16; // output column
    const int n0 = chunk * 256;

    v8f c = {0.f, 0.f, 0.f, 0.f, 0.f, 0.f, 0.f, 0.f};
    for (int nn = 0; nn < 256; nn += 4) {
        const int n = n0 + nn + 2 * half;
        v2f a;  // A = attn.T : A[slot=l16][K=n]
        a.x = attn[(size_t)n * K_SLOTS + l16];
        a.y = attn[(size_t)(n + 1) * K_SLOTS + l16];
        v2f b;  // B = v : B[K=n][h]
        b.x = vmat[(size_t)n * H_DIM + h];
        b.y = vmat[(size_t)(n + 1) * H_DIM + h];
        c = wmma4(a, b, c);
    }
    #pragma unroll
    for (int j = 0; j < 8; ++j)
        up_part[(size_t)chunk * (K_SLOTS * H_DIM) + (j + 8 * half) * H_DIM + h] = c[j];
}

// Deterministic reduction of update partials: grid=16, block=256
__global__ void k_reduce_up(const float* __restrict__ up_part, float* __restrict__ updates) {
    const int idx = blockIdx.x * 256 + threadIdx.x;  // 0..4095
    float s = 0.f;
    for (int ch = 0; ch < UP_CHUNKS; ++ch) s += up_part[(size_t)ch * (K_SLOTS * H_DIM) + idx];
    updates[idx] = s;
}

// ---------------------------------------------------------------------------
// Kernel: GRU cell + LN + MLP on [16,256] state. 1 block, 512 threads.
// Also finalizes the attention-variance statistic for this iteration.
// ---------------------------------------------------------------------------
__global__ void k_gru_mlp(const float* __restrict__ updates, const float* __restrict__ colsum,
                          const float* __restrict__ sqsum,
                          float* __restrict__ slots,
                          const float* __restrict__ W_ih, const float* __restrict__ W_hh,
                          const float* __restrict__ b_ih, const float* __restrict__ b_hh,
                          const float* __restrict__ lnw, const float* __restrict__ lnb,
                          const float* __restrict__ w1, const float* __restrict__ b1,
                          const float* __restrict__ w2, const float* __restrict__ b2,
                          float* __restrict__ varpart, int iter, float* __restrict__ out) {
    __shared__ float smem[12288];           // 48 KB
    float* u  = smem;                        // [16][256] scaled updates
    float* hp = smem + 4096;                 // [16][256] prev slots
    float* r  = smem + 8192;                 // [16][256] gate r, then hnew
    const int tid = threadIdx.x;

    // variance of normalized attention (mean is exactly 1/N analytically)
    if (tid == 0) {
        float s2 = 0.f;
        for (int s = 0; s < K_SLOTS; ++s) {
            float cs = colsum[s];
            s2 += sqsum[s] / (cs * cs);
        }
        const float nk = (float)N_IN * (float)K_SLOTS;
        float vp = (s2 - nk * (1.0f / ((float)N_IN * (float)N_IN))) / (nk - 1.0f);
        varpart[iter] = vp;
        if (iter == 2) out[K_SLOTS * H_DIM] = (varpart[0] + varpart[1] + vp) * (1.0f / 3.0f);
    }

    for (int idx = tid; idx < 4096; idx += 512) {
        u[idx]  = updates[idx] / colsum[idx >> 8];
        hp[idx] = slots[idx];
    }
    __syncthreads();

    // gate r
    for (int idx = tid; idx < 4096; idx += 512) {
        const int s = idx >> 8, j = idx & 255;
        const float* wi = W_ih + (size_t)j * H_DIM;           // row j of [768,256]
        const float* wh = W_hh + (size_t)j * H_DIM;
        float gr = b_ih[j], hr = b_hh[j];
        for (int d = 0; d < H_DIM; ++d) {
            gr += u[s * H_DIM + d]  * wi[d];
            hr += hp[s * H_DIM + d] * wh[d];
        }
        r[idx] = sigmoidf_(gr + hr);
    }
    __syncthreads();

    // gates z, n -> hnew (global slots + LDS r)
    for (int idx = tid; idx < 4096; idx += 512) {
        const int s = idx >> 8, j = idx & 255;
        const float* wiz = W_ih + (size_t)(256 + j) * H_DIM;
        const float* whz = W_hh + (size_t)(256 + j) * H_DIM;
        const float* win = W_ih + (size_t)(512 + j) * H_DIM;
        const float* whn = W_hh + (size_t)(512 + j) * H_DIM;
        float gz = b_ih[256 + j], hz = b_hh[256 + j];
        float gn = b_ih[512 + j], hn = b_hh[512 + j];
        for (int d = 0; d < H_DIM; ++d) {
            const float uv = u[s * H_DIM + d], hv = hp[s * H_DIM + d];
            gz += uv * wiz[d]; hz += hv * whz[d];
            gn += uv * win[d]; hn += hv * whn[d];
        }
        const float z = sigmoidf_(gz + hz);
        const float n = tanhf(gn + r[idx] * hn);
        const float hnew = (1.0f - z) * n + z * hp[idx];
        slots[idx] = hnew;   // residual source
        r[idx] = hnew;       // for LN
    }
    __syncthreads();

    // LN(hnew) -> u (one wave per row)
    {
        const int row = tid >> 5, lane = tid & 31;
        float s = 0.f, sq = 0.f;
        for (int d = lane; d < H_DIM; d += 32) { float t = r[row * H_DIM + d]; s += t; sq += t * t; }
        for (int mk = 16; mk >= 1; mk >>= 1) { s += __shfl_xor(s, mk, 32); sq += __shfl_xor(sq, mk, 32); }
        const float mu = s * (1.0f / H_DIM);
        const float rs = rsqrtf(sq * (1.0f / H_DIM) - mu * mu + LN_EPS);
        for (int d = lane; d < H_DIM; d += 32)
            u[row * H_DIM + d] = (r[row * H_DIM + d] - mu) * rs * lnw[d] + lnb[d];
    }
    __syncthreads();

    // h1 = relu(ln @ w1.T + b1)  [16,512] -> smem+4096 .. +12287
    float* h1 = smem + 4096;
    for (int idx = tid; idx < 8192; idx += 512) {
        const int s = idx >> 9, m = idx & 511;
        const float* wr = w1 + (size_t)m * H_DIM;  // [512,256]
        float acc = b1[m];
        for (int d = 0; d < H_DIM; ++d) acc += u[s * H_DIM + d] * wr[d];
        h1[idx] = fmaxf(acc, 0.0f);
    }
    __syncthreads();

    // slots += h1 @ w2.T + b2
    for (int idx = tid; idx < 4096; idx += 512) {
        const int s = idx >> 8, h = idx & 255;
        const float* wr = w2 + (size_t)h * D_MLP;  // [256,512]
        float acc = b2[h];
        for (int m = 0; m < D_MLP; ++m) acc += h1[s * D_MLP + m] * wr[m];
        const float res = slots[idx] + acc;
        slots[idx] = res;
        if (iter == 2) out[idx] = res;
    }
}

// ---------------------------------------------------------------------------
extern "C" void kernel_launch(void* const* d_in, const int* in_sizes, int n_in,
                              void* d_out, int out_size, void* d_ws, size_t ws_size,
                              hipStream_t stream) {
    (void)in_sizes; (void)n_in; (void)out_size; (void)ws_size;
    const float* inputs     = (const float*)d_in[0];
    const float* ln_in_w    = (const float*)d_in[1];
    const float* ln_in_b    = (const float*)d_in[2];
    const float* ln_slots_w = (const float*)d_in[3];
    const float* ln_slots_b = (const float*)d_in[4];
    const float* ln_mlp_w   = (const float*)d_in[5];
    const float* ln_mlp_b   = (const float*)d_in[6];
    const float* slots_init = (const float*)d_in[7];
    const float* Wq         = (const float*)d_in[8];
    const float* Wk         = (const float*)d_in[9];
    const float* Wv         = (const float*)d_in[10];
    const float* W_ih       = (const float*)d_in[11];
    const float* W_hh       = (const float*)d_in[12];
    const float* b_ih       = (const float*)d_in[13];
    const float* b_hh       = (const float*)d_in[14];
    const float* mlp_w1     = (const float*)d_in[15];
    const float* mlp_b1     = (const float*)d_in[16];
    const float* mlp_w2     = (const float*)d_in[17];
    const float* mlp_b2     = (const float*)d_in[18];
    float* out = (float*)d_out;

    // workspace carve-up (floats)
    float* W = (float*)d_ws;
    float* kbuf    = W;                               // 16384*256
    float* vbuf    = kbuf + (size_t)N_IN * H_DIM;     // 16384*256
    float* attn    = vbuf + (size_t)N_IN * H_DIM;     // 16384*16
    float* qbuf    = attn + (size_t)N_IN * K_SLOTS;   // 4096
    float* slots   = qbuf + K_SLOTS * H_DIM;          // 4096
    float* cs_part = slots + K_SLOTS * H_DIM;         // 1024*16
    float* sq_part = cs_part + N_TILES * K_SLOTS;     // 1024*16
    float* colsum  = sq_part + N_TILES * K_SLOTS;     // 16
    float* sqsum   = colsum + K_SLOTS;                // 16
    float* up_part = sqsum + K_SLOTS;                 // 64*4096
    float* updates = up_part + (size_t)UP_CHUNKS * K_SLOTS * H_DIM;  // 4096
    float* varpart = updates + K_SLOTS * H_DIM;       // 3

    k_ln_kv<<<N_TILES, 256, 0, stream>>>(inputs, ln_in_w, ln_in_b, Wk, Wv, kbuf, vbuf);
    k_init<<<1, 256, 0, stream>>>(slots_init, slots);

    for (int it = 0; it < 3; ++it) {
        k_q<<<1, 512, 0, stream>>>(slots, ln_slots_w, ln_slots_b, Wq, qbuf);
        k_attn<<<N_TILES, 32, 0, stream>>>(kbuf, qbuf, attn, cs_part, sq_part);
        k_reduce_cs<<<1, 32, 0, stream>>>(cs_part, sq_part, colsum, sqsum);
        k_updates<<<dim3(UP_CHUNKS, 16), 32, 0, stream>>>(attn, vbuf, up_part);
        k_reduce_up<<<16, 256, 0, stream>>>(up_part, updates);
        k_gru_mlp<<<1, 512, 0, stream>>>(updates, colsum, sqsum, slots,
                                         W_ih, W_hh, b_ih, b_hh,
                                         ln_mlp_w, ln_mlp_b,
                                         mlp_w1, mlp_b1, mlp_w2, mlp_b2,
                                         varpart, it, out);
    }
}